// OnlineReweightingLoss_71244917506325
// MI455X (gfx1250) — compile-verified
//
#include <hip/hip_runtime.h>
#include <math.h>

// Problem constants (fixed by the reference)
#define CLASSES        64
#define SGROUPS        8
#define NKEYS          (CLASSES * SGROUPS)      // 512
#define ROW_BYTES      (CLASSES * 4)            // 256
#define TB_MAIN        128                      // 4 waves of 32
#define WAVES_PER_BLK  (TB_MAIN / 32)
#define ROWS_PER_CHUNK 32                       // one row per lane
#define LDS_ROW_BYTES  272                      // 256 + 16 pad -> dword stride 68 (bank stride 4)
#define BLOCKS_MAIN    1024

// ---------------- CDNA5 async global->LDS path (guarded) ----------------
#if __has_builtin(__builtin_amdgcn_global_load_async_to_lds_b128) && \
    __has_builtin(__builtin_amdgcn_s_wait_asynccnt)
#define USE_ASYNC_LDS 1
#endif

typedef int v4i __attribute__((vector_size(16)));   // matches builtin param type

__device__ __forceinline__ void async_copy_b128(void* lds_dst, const void* g_src) {
#if defined(USE_ASYNC_LDS)
  __builtin_amdgcn_global_load_async_to_lds_b128(
      (__attribute__((address_space(1))) v4i*)(g_src),
      (__attribute__((address_space(3))) v4i*)(lds_dst),
      0, 0);
#else
  *(float4*)lds_dst = *(const float4*)g_src;
#endif
}

__device__ __forceinline__ void async_wait_all() {
#if defined(USE_ASYNC_LDS)
  __builtin_amdgcn_s_wait_asynccnt(0);
#endif
}

// ---------------- exact-f32 wave32 sum via V_WMMA_F32_16X16X4_F32 ----------------
typedef float v2f __attribute__((ext_vector_type(2)));
typedef float v8f __attribute__((ext_vector_type(8)));

__device__ __forceinline__ float wave_reduce_sum_f32(float v) {
#if __has_builtin(__builtin_amdgcn_wmma_f32_16x16x4_f32)
  // A(16x4): lane m<16 -> {A[m][0],A[m][1]}, lane m+16 -> {A[m][2],A[m][3]}
  // With a = (v, 0) and B = ones(4x16):  D[m][n] = v_m + v_{m+16}
  v2f a; a[0] = v;    a[1] = 0.0f;
  v2f b; b[0] = 1.0f; b[1] = 1.0f;
  v8f c = {0.f, 0.f, 0.f, 0.f, 0.f, 0.f, 0.f, 0.f};
  v8f d = __builtin_amdgcn_wmma_f32_16x16x4_f32(false, a, false, b, (short)0, c,
                                                false, false);
  // lane<16: sum_{r=0..7} D[r][n]   = v_0..7  + v_16..23
  // lane>=16: sum_{r=0..7} D[r+8][n] = v_8..15 + v_24..31
  float s = ((d[0] + d[1]) + (d[2] + d[3])) + ((d[4] + d[5]) + (d[6] + d[7]));
  s += __shfl_down(s, 16, 32);   // lane0 now holds the full 32-lane sum
  return s;
#else
  #pragma unroll
  for (int off = 16; off > 0; off >>= 1) v += __shfl_down(v, off, 32);
  return v;
#endif
}

// ---------------- kernel 0: zero the counts table ----------------
__global__ void OnlineRW_zero_kernel(unsigned* counts) {
  int i = threadIdx.x;
  if (i < NKEYS) counts[i] = 0u;
}

// ---------------- kernel 1: (target, subgroup) histogram ----------------
__global__ void OnlineRW_hist_kernel(const int* __restrict__ tgt,
                                     const int* __restrict__ sg,
                                     unsigned* __restrict__ counts, int n) {
  __shared__ unsigned h[NKEYS];
  for (int i = threadIdx.x; i < NKEYS; i += blockDim.x) h[i] = 0u;
  __syncthreads();
  const int stride = gridDim.x * blockDim.x;
  for (int i = blockIdx.x * blockDim.x + threadIdx.x; i < n; i += stride)
    atomicAdd(&h[tgt[i] * SGROUPS + sg[i]], 1u);
  __syncthreads();
  for (int i = threadIdx.x; i < NKEYS; i += blockDim.x)
    if (h[i]) atomicAdd(&counts[i], h[i]);
}

// ---------------- kernel 2: streaming CE + reweight + block partials ----------------
__global__ __launch_bounds__(TB_MAIN) void OnlineRW_loss_kernel(
    const float* __restrict__ logits, const int* __restrict__ tgt,
    const int* __restrict__ sg, const unsigned* __restrict__ counts,
    float* __restrict__ partials, int n) {
  __shared__ __align__(16) unsigned char smem[WAVES_PER_BLK * ROWS_PER_CHUNK * LDS_ROW_BYTES];
  __shared__ float cntTab[NKEYS];
  __shared__ float wsum[WAVES_PER_BLK];

  for (int i = threadIdx.x; i < NKEYS; i += blockDim.x)
    cntTab[i] = (float)counts[i];
  __syncthreads();

  const int lane   = threadIdx.x & 31;
  const int wid    = threadIdx.x >> 5;
  const int gwave  = blockIdx.x * WAVES_PER_BLK + wid;
  const int nwaves = gridDim.x * WAVES_PER_BLK;

  unsigned char* wbase = smem + wid * (ROWS_PER_CHUNK * LDS_ROW_BYTES);
  // coalesced staging: async instr j moves rows {2j, 2j+1}; lane L covers 16B
  unsigned char* ldsDst0 = wbase + (lane >> 4) * LDS_ROW_BYTES + (lane & 15) * 16;
  const unsigned char* myRow = wbase + lane * LDS_ROW_BYTES;

  float acc = 0.0f;
  const int fullChunks = n >> 5;

  for (int ch = gwave; ch < fullChunks; ch += nwaves) {
    const unsigned char* g0 = (const unsigned char*)logits +
                              (size_t)ch * (ROWS_PER_CHUNK * ROW_BYTES) +
                              (size_t)lane * 16;
    #pragma unroll
    for (int j = 0; j < 16; ++j)
      async_copy_b128(ldsDst0 + j * (2 * LDS_ROW_BYTES), g0 + j * 512);
    async_wait_all();

    // lane-private row: no cross-lane ops in the hot loop
    float4 x[16];
    #pragma unroll
    for (int i = 0; i < 16; ++i)
      x[i] = *(const float4*)(myRow + i * 16);

    float m = x[0].x;
    #pragma unroll
    for (int i = 0; i < 16; ++i)
      m = fmaxf(m, fmaxf(fmaxf(x[i].x, x[i].y), fmaxf(x[i].z, x[i].w)));

    float s = 0.0f;
    #pragma unroll
    for (int i = 0; i < 16; ++i) {
      s += __expf(x[i].x - m);
      s += __expf(x[i].y - m);
      s += __expf(x[i].z - m);
      s += __expf(x[i].w - m);
    }

    const int row = (ch << 5) + lane;
    const int t = tgt[row];
    const int g = sg[row];
    const float xt = *(const float*)(myRow + t * 4);
    const float loss = (m - xt) + __logf(s);
    acc += loss / cntTab[t * SGROUPS + g];
  }

  // tail rows (n % 32 != 0): wave 0, direct global loads, lane-guarded
  const int tailStart = fullChunks << 5;
  if (gwave == 0 && tailStart < n) {
    const int row = tailStart + lane;
    if (row < n) {
      const float4* rp = (const float4*)(logits + (size_t)row * CLASSES);
      float4 y[16];
      #pragma unroll
      for (int i = 0; i < 16; ++i) y[i] = rp[i];
      float m = y[0].x;
      #pragma unroll
      for (int i = 0; i < 16; ++i)
        m = fmaxf(m, fmaxf(fmaxf(y[i].x, y[i].y), fmaxf(y[i].z, y[i].w)));
      float s = 0.0f;
      #pragma unroll
      for (int i = 0; i < 16; ++i) {
        s += __expf(y[i].x - m);
        s += __expf(y[i].y - m);
        s += __expf(y[i].z - m);
        s += __expf(y[i].w - m);
      }
      const int t = tgt[row];
      const int g = sg[row];
      const float xt = ((const float*)rp)[t];
      acc += ((m - xt) + __logf(s)) / cntTab[t * SGROUPS + g];
    }
  }

  // uniform control flow here: EXEC is all-ones for the WMMA reduction
  const float wtot = wave_reduce_sum_f32(acc);
  if (lane == 0) wsum[wid] = wtot;
  __syncthreads();
  if (threadIdx.x == 0) {
    float b = 0.0f;
    #pragma unroll
    for (int w = 0; w < WAVES_PER_BLK; ++w) b += wsum[w];
    partials[blockIdx.x] = b;
  }
}

// ---------------- kernel 3: deterministic final reduction (1 wave) ----------------
__global__ void OnlineRW_final_kernel(const float* __restrict__ partials, int nPart,
                                      float* __restrict__ out) {
  const int lane = threadIdx.x;  // launched with exactly 32 threads
  float s = 0.0f;
  for (int i = lane; i < nPart; i += 32) s += partials[i];
  const float tot = wave_reduce_sum_f32(s);
  if (lane == 0) out[0] = tot;
}

extern "C" void kernel_launch(void* const* d_in, const int* in_sizes, int n_in,
                              void* d_out, int out_size, void* d_ws, size_t ws_size,
                              hipStream_t stream) {
  (void)n_in; (void)out_size; (void)ws_size;
  const float* logits = (const float*)d_in[0];
  const int*   tgt    = (const int*)d_in[1];
  const int*   sg     = (const int*)d_in[2];
  const int    n      = in_sizes[1];

  // ws layout: [0, 2KB) u32 counts[512]; [2KB, 2KB+4*BLOCKS_MAIN) f32 partials
  unsigned* counts   = (unsigned*)d_ws;
  float*    partials = (float*)((char*)d_ws + 2048);
  float*    out      = (float*)d_out;

  OnlineRW_zero_kernel<<<1, 512, 0, stream>>>(counts);
  OnlineRW_hist_kernel<<<256, 256, 0, stream>>>(tgt, sg, counts, n);
  OnlineRW_loss_kernel<<<BLOCKS_MAIN, TB_MAIN, 0, stream>>>(logits, tgt, sg, counts,
                                                            partials, n);
  OnlineRW_final_kernel<<<1, 32, 0, stream>>>(partials, BLOCKS_MAIN, out);
}